// Dis_61555471286909
// MI455X (gfx1250) — compile-verified
//
#include <hip/hip_runtime.h>
#include <hip/hip_bf16.h>

// LSTM forward, MI455X (gfx1250), wave32.
// Fused recurrence: gates = [x_t | h] @ [W_ih|W_hh]^T + (b_ih+b_hh), K = 32+64 = 96.
// One workgroup per 16 batch rows (grid = 32), persistent over all T=512 steps.
// f32 WMMA 16x16x4 keeps full fp32 precision through the 512-step scan.
// Weights live in registers (96 VGPR B-fragments per lane) for the whole kernel.
// x tile is register double-buffered: the global load for step t+1 issues at the
// top of step t, hiding HBM/L2 latency behind the whole step's compute.

typedef __attribute__((ext_vector_type(2))) float v2f;
typedef __attribute__((ext_vector_type(4))) float v4f;
typedef __attribute__((ext_vector_type(8))) float v8f;

namespace {
constexpr int B = 512, T = 512, D = 32, H = 64;
constexpr int KC = D + H;       // 96 concatenated K
constexpr int BM = 16;          // batch rows per workgroup (WMMA M)
constexpr int KSTEPS = KC / 4;  // 24 f32 WMMA K-steps
constexpr int CATS = 100;       // cat row stride (pad: 36m mod 64 distinct -> conflict free)
constexpr int GATS = 260;       // gates row stride (pad: lane halves land 32 banks apart)
}

__global__ __launch_bounds__(256, 1)
void lstm_scan_kernel(const float* __restrict__ x,
                      const float* __restrict__ h0,
                      const float* __restrict__ c0,
                      const float* __restrict__ Wih,
                      const float* __restrict__ Whh,
                      const float* __restrict__ bih,
                      const float* __restrict__ bhh,
                      const float* __restrict__ Wout,
                      const float* __restrict__ bout,
                      float* __restrict__ out_head,  // [B,T]
                      float* __restrict__ out_h)     // [B,T,H]
{
    __shared__ float cat[BM * CATS];   // [m][0:32]=x_t, [m][32:96]=h_t
    __shared__ float cst[BM * H];      // cell state c
    __shared__ float gates[BM * GATS]; // 16 x 256 gate exchange

    const int tid  = threadIdx.x;
    const int lane = tid & 31;
    const int wave = tid >> 5;
    const int half = lane >> 4;        // 0: lanes 0-15, 1: lanes 16-31
    const int lm   = lane & 15;        // M (A/C rows) or N (B/C cols) in tile
    const int b0   = blockIdx.x * BM;

    // ---- Preload per-wave B fragments (weights) into registers, once. ----
    // Wave w covers gate columns n0 = 32w .. 32w+31 as two 16-wide N tiles.
    // B 4x16 f32 layout (symmetric to A): vgpr0 = row k0 (lanes 0-15) /
    // row k0+2 (lanes 16-31); vgpr1 = rows k0+1 / k0+3.
    v2f Bfrag[2][KSTEPS];
#pragma unroll
    for (int nt = 0; nt < 2; ++nt) {
        const int n = wave * 32 + nt * 16 + lm;  // gate column g
#pragma unroll
        for (int ks = 0; ks < KSTEPS; ++ks) {
            const int kk = ks * 4 + half * 2;    // even, never straddles D=32
            const float* src = (kk < D) ? (Wih + n * D + kk)
                                        : (Whh + n * H + (kk - D));
            Bfrag[nt][ks] = *(const v2f*)src;    // global_load_b64
        }
    }
    const float bias0 = bih[wave * 32 + lm]      + bhh[wave * 32 + lm];
    const float bias1 = bih[wave * 32 + 16 + lm] + bhh[wave * 32 + 16 + lm];
    const float bo = bout[0];

    // Head weights: each 16-lane half holds W_out[lm*4 .. lm*4+3] (4 regs).
    const v4f wreg = *(const v4f*)(Wout + lm * 4);

    // ---- Init h, c state in LDS ----
    for (int i = tid; i < BM * H; i += 256) {
        const int m = i >> 6, j = i & 63;
        cat[m * CATS + D + j] = h0[(size_t)(b0 + m) * H + j];
        cst[m * H + j]        = c0[(size_t)(b0 + m) * H + j];
    }

    // ---- x tile register double buffer: thread owns (row xm, cols xd..xd+1) ----
    const int xm = tid >> 4;
    const int xd = (tid & 15) * 2;
    const float* xbase = x + (size_t)(b0 + xm) * T * D + xd;
    v2f xreg = *(const v2f*)xbase;     // prefetch t = 0

    for (int t = 0; t < T; ++t) {
        __syncthreads();
        // ---- Stage x_t from registers; immediately issue load for t+1 ----
        cat[xm * CATS + xd]     = xreg.x;
        cat[xm * CATS + xd + 1] = xreg.y;
        {
            const int tn = (t + 1 < T) ? (t + 1) : (T - 1);  // branchless tail
            xreg = *(const v2f*)(xbase + (size_t)tn * D);    // hidden by step t
        }
        __syncthreads();

        // ---- gates(16x256) += cat(16x96) x W^T(96x256), fp32 WMMA ----
        v8f acc0 = {}; v8f acc1 = {};
#pragma unroll
        for (int ks = 0; ks < KSTEPS; ++ks) {
            const int kk = ks * 4 + half * 2;
            // A 16x4 f32: vgpr0 = K=k0 (lanes 0-15) / k0+2 (16-31); vgpr1 = +1
            v2f a = *(const v2f*)(&cat[lm * CATS + kk]);  // ds_load_2addr_b64 pairs
            acc0 = __builtin_amdgcn_wmma_f32_16x16x4_f32(
                false, a, false, Bfrag[0][ks], (short)0, acc0, false, false);
            acc1 = __builtin_amdgcn_wmma_f32_16x16x4_f32(
                false, a, false, Bfrag[1][ks], (short)0, acc1, false, false);
        }
        // C/D layout: vgpr r -> M = r + 8*half, N = lane%16
#pragma unroll
        for (int r = 0; r < 8; ++r) {
            const int m = r + half * 8;
            gates[m * GATS + wave * 32 + lm]      = acc0[r] + bias0;
            gates[m * GATS + wave * 32 + 16 + lm] = acc1[r] + bias1;
        }
        __syncthreads();

        // ---- Elementwise LSTM cell: 16*64 items over 256 threads ----
#pragma unroll
        for (int r = 0; r < 4; ++r) {
            const int item = tid + r * 256;
            const int m = item >> 6, j = item & 63;
            const float gi = gates[m * GATS + j];
            const float gf = gates[m * GATS + 64 + j];
            const float gg = gates[m * GATS + 128 + j];
            const float go = gates[m * GATS + 192 + j];
            const float si = 1.0f / (1.0f + __expf(-gi));
            const float sf = 1.0f / (1.0f + __expf(-gf));
            const float so = 1.0f / (1.0f + __expf(-go));
            const float cv = sf * cst[m * H + j] + si * tanhf(gg);
            const float hv = so * tanhf(cv);
            cst[m * H + j] = cv;
            cat[m * CATS + D + j] = hv;                          // feeds step t+1
            out_h[((size_t)(b0 + m) * T + t) * H + j] = hv;      // [B,T,H] output
        }
        __syncthreads();

        // ---- Head: output[b][t] = sigmoid(h . W_out + b_out) ----
        // Wave w reduces rows 2w (lanes 0-15) and 2w+1 (lanes 16-31):
        // each lane does a 4-wide partial dot, then a 4-step shfl_xor butterfly
        // (masks < 16 keep the two halves independent on wave32).
        {
            const int row = 2 * wave + half;
            v4f hv4 = *(const v4f*)(&cat[row * CATS + D + lm * 4]); // ds_load_b128
            float acc = hv4.x * wreg.x + hv4.y * wreg.y +
                        hv4.z * wreg.z + hv4.w * wreg.w;
            acc += __shfl_xor(acc, 1, 32);
            acc += __shfl_xor(acc, 2, 32);
            acc += __shfl_xor(acc, 4, 32);
            acc += __shfl_xor(acc, 8, 32);
            if (lm == 0)
                out_head[(size_t)(b0 + row) * T + t] =
                    1.0f / (1.0f + __expf(-(acc + bo)));
        }
    }
}

extern "C" void kernel_launch(void* const* d_in, const int* in_sizes, int n_in,
                              void* d_out, int out_size, void* d_ws, size_t ws_size,
                              hipStream_t stream) {
    (void)in_sizes; (void)n_in; (void)d_ws; (void)ws_size; (void)out_size;
    const float* x    = (const float*)d_in[0];
    const float* h0   = (const float*)d_in[1];
    const float* c0   = (const float*)d_in[2];
    const float* Wih  = (const float*)d_in[3];
    const float* Whh  = (const float*)d_in[4];
    const float* bih  = (const float*)d_in[5];
    const float* bhh  = (const float*)d_in[6];
    const float* Wout = (const float*)d_in[7];
    const float* bout = (const float*)d_in[8];
    // d_out = [output (B*T floats)] ++ [hs (B*T*H floats)]
    float* out_head = (float*)d_out;
    float* out_h    = (float*)d_out + (size_t)B * T;

    lstm_scan_kernel<<<B / BM, 256, 0, stream>>>(
        x, h0, c0, Wih, Whh, bih, bhh, Wout, bout, out_head, out_h);
}